// YOLOv1Loss_32134945309064
// MI455X (gfx1250) — compile-verified
//
#include <hip/hip_runtime.h>
#include <hip/hip_bf16.h>

// YOLOv1 loss, MI455X (gfx1250). Memory-bound: ~96MB streamed once -> ~4.1us
// at 23.3 TB/s. Strategy: 2 cells (240B, 16B-aligned) per thread via b128
// loads, fp32 math, WMMA-based wave reduction, LDS block stage, one
// global_atomic_add_f32 per block.

typedef float v2f __attribute__((ext_vector_type(2)));
typedef float v8f __attribute__((ext_vector_type(8)));

__device__ __forceinline__ float cell_loss(const float* __restrict__ p,
                                           const float* __restrict__ t,
                                           float fi, float fj) {
  const float cw = 1.0f / 7.0f;
  const float obj = (t[8] > 0.0f) ? 1.0f : 0.0f;

  // Target box 0 corners (reference uses only the first target box)
  const float txc = (t[0] + fi) * cw;
  const float tyc = (t[1] + fj) * cw;
  const float tx1 = txc - 0.5f * t[2], ty1 = tyc - 0.5f * t[3];
  const float tx2 = txc + 0.5f * t[2], ty2 = tyc + 0.5f * t[3];
  const float areaT = (tx2 - tx1) * (ty2 - ty1);

  float iou0 = 0.0f, iou1 = 0.0f;
#pragma unroll
  for (int b = 0; b < 2; ++b) {
    const float bx = p[4 * b + 0], by = p[4 * b + 1];
    const float bw = p[4 * b + 2], bh = p[4 * b + 3];
    const float xc = (bx + fi) * cw, yc = (by + fj) * cw;
    const float x1 = xc - 0.5f * bw, y1 = yc - 0.5f * bh;
    const float x2 = xc + 0.5f * bw, y2 = yc + 0.5f * bh;
    const float lx = fmaxf(x1, tx1), ly = fmaxf(y1, ty1);
    const float rx = fminf(x2, tx2), ry = fminf(y2, ty2);
    const float iw = fmaxf(rx - lx, 0.0f), ih = fmaxf(ry - ly, 0.0f);
    const float inter = iw * ih;
    const float areaP = (x2 - x1) * (y2 - y1);
    const float iou = inter / (areaP + areaT - inter);
    if (b == 0) iou0 = iou; else iou1 = iou;
  }

  // argmax over B with first-index tie-break (box 0 on ties)
  const bool r1 = (iou1 > iou0);
  const float pc0 = p[8], pc1 = p[9];
  const float rc = r1 ? pc1 : pc0;   // responsible conf
  const float oc = r1 ? pc0 : pc1;   // non-responsible conf

  const float l_resp  = (rc - 1.0f) * (rc - 1.0f);
  const float l_nore  = oc * oc;

  float c0 = 0.0f, c1 = 0.0f;
#pragma unroll
  for (int k = 0; k < 4; ++k) {
    const float d0 = p[k] - t[k];         c0 += d0 * d0;
    const float d1 = p[4 + k] - t[4 + k]; c1 += d1 * d1;
  }
  const float l_coord = r1 ? c1 : c0;

  float l_cls = 0.0f;
#pragma unroll
  for (int k = 0; k < 20; ++k) {
    const float d = p[10 + k] - t[10 + k];
    l_cls += d * d;
  }

  const float l_noobj = pc0 * pc0 + pc1 * pc1;

  // 5*coord + 1*resp + 0.5*noresp + 1*cls (obj cells); 0.5*noobj (empty cells)
  return obj * (5.0f * l_coord + l_resp + 0.5f * l_nore + l_cls)
       + (1.0f - obj) * (0.5f * l_noobj);
}

__global__ void __launch_bounds__(256)
yolo_loss_kernel(const float* __restrict__ pred, const float* __restrict__ tgt,
                 float* __restrict__ out, int nPairs, float invN) {
  const int tid = blockIdx.x * blockDim.x + threadIdx.x;

  float partial = 0.0f;
  if (tid < nPairs) {
    // 2 cells = 60 floats = 240 bytes, base is 16B-aligned -> 15 b128 loads each
    const float4* pv = reinterpret_cast<const float4*>(pred) + (size_t)tid * 15;
    const float4* tv = reinterpret_cast<const float4*>(tgt)  + (size_t)tid * 15;
    float P[60], T[60];
#pragma unroll
    for (int q = 0; q < 15; ++q) {
      const float4 v = pv[q];
      P[4 * q + 0] = v.x; P[4 * q + 1] = v.y; P[4 * q + 2] = v.z; P[4 * q + 3] = v.w;
    }
#pragma unroll
    for (int q = 0; q < 15; ++q) {
      const float4 v = tv[q];
      T[4 * q + 0] = v.x; T[4 * q + 1] = v.y; T[4 * q + 2] = v.z; T[4 * q + 3] = v.w;
    }

    const int cell0 = tid * 2;
    const int m0 = cell0 % 49;
    const int m1 = (cell0 + 1) % 49;  // pair may cross the 49-cell image boundary
    partial  = cell_loss(P,      T,      (float)(m0 / 7), (float)(m0 % 7));
    partial += cell_loss(P + 30, T + 30, (float)(m1 / 7), (float)(m1 % 7));
  }

  // --- Wave32 reduction via V_WMMA_F32_16X16X4_F32 ---
  // A (16x4): lanes 0-15 VGPR0 hold K=0 (their partial), lanes 16-31 K=2;
  // VGPR1 (K=1/K=3) = 0.  B (4x16) = all ones.  D[m][n] = sum_k A[m][k].
  // Lane n (n<16) then holds rows m=0..7 across its 8 D VGPRs, lane 16+n rows
  // m=8..15; summing the 8 VGPRs + one xor-16 shuffle yields the wave total.
  v2f a;  a.x = partial; a.y = 0.0f;
  v2f bo; bo.x = 1.0f;   bo.y = 1.0f;
  v8f c = {0.0f, 0.0f, 0.0f, 0.0f, 0.0f, 0.0f, 0.0f, 0.0f};
  v8f d = __builtin_amdgcn_wmma_f32_16x16x4_f32(
      /*neg_a=*/false, a, /*neg_b=*/false, bo,
      /*c_mod=*/(short)0, c, /*reuse_a=*/false, /*reuse_b=*/false);
  float s = d[0] + d[1] + d[2] + d[3] + d[4] + d[5] + d[6] + d[7];
  s += __shfl_xor(s, 16, 32);  // combine half-wave row groups -> full wave sum

  // --- Block reduction: 8 waves -> LDS -> 1 atomic per block ---
  __shared__ float wsum[8];
  const int lane = threadIdx.x & 31;
  const int wid  = threadIdx.x >> 5;
  if (lane == 0) wsum[wid] = s;
  __syncthreads();
  if (threadIdx.x == 0) {
    float bs = 0.0f;
#pragma unroll
    for (int w = 0; w < 8; ++w) bs += wsum[w];
    atomicAdd(out, bs * invN);
  }
}

__global__ void zero_out_kernel(float* out) { out[0] = 0.0f; }

extern "C" void kernel_launch(void* const* d_in, const int* in_sizes, int n_in,
                              void* d_out, int out_size, void* d_ws, size_t ws_size,
                              hipStream_t stream) {
  const float* pred = (const float*)d_in[0];
  const float* tgt  = (const float*)d_in[1];
  float* out = (float*)d_out;

  const long totalCells = (long)in_sizes[0] / 30;   // N*S*S = 401408
  const int  nPairs     = (int)(totalCells / 2);    // even: 200704
  const float invN      = 1.0f / (float)(totalCells / 49);  // 1/N

  zero_out_kernel<<<1, 1, 0, stream>>>(out);
  const int grid = (nPairs + 255) / 256;
  yolo_loss_kernel<<<grid, 256, 0, stream>>>(pred, tgt, out, nPairs, invN);
}